// MutualAttention_43499428773891
// MI455X (gfx1250) — compile-verified
//
#include <hip/hip_runtime.h>
#include <hip/hip_bf16.h>

typedef __attribute__((ext_vector_type(2))) float v2f;
typedef __attribute__((ext_vector_type(8))) float v8f;

#define DIM   1024
#define BSZ   32
#define LEN   512
#define TOPK  256

// ---------------------------------------------------------------------------
// Kernel 1: fused  X = H1_tile @ W  (into LDS)  then  S = X @ H2^T, epilogue
// writes order-preserving uint32 keys (invalid pairs -> 0) to workspace.
// grid = (32 i-tiles, 32 batches), block = 256 threads (8 waves).
// ---------------------------------------------------------------------------
__global__ __launch_bounds__(256) void bilinear_keys_kernel(
    const float* __restrict__ in1,   // [L1, B, D]
    const float* __restrict__ in2,   // [L2, B, D]
    const int*   __restrict__ m1,    // [L1, B]
    const int*   __restrict__ m2,    // [L2, B]
    const float* __restrict__ Wm,    // [D, D]
    const float* __restrict__ bias,  // [1]
    unsigned*    __restrict__ keys)  // [B, L1, L2]
{
    __shared__ float Xs[16 * DIM];   // 64 KB X tile, additively swizzled

    const int it   = blockIdx.x;          // i-tile 0..31
    const int bb   = blockIdx.y;          // batch 0..31
    const int i0   = it * 16;
    const int lane = threadIdx.x & 31;
    const int w    = threadIdx.x >> 5;    // wave id 0..7
    const int lm   = lane & 15;           // row (A) / col (B,C) within tile
    const int hi   = lane >> 4;           // 0 or 1
    const int koffs = hi * 2;             // A/B K sub-offset per ISA layout

    // -------- Stage 1: Xs(16x1024) = H1[b, i0:i0+16, :] @ W --------
    v8f acc[8] = {};
    const float* arow = in1 + ((size_t)(i0 + lm) * BSZ + bb) * DIM + koffs;

    for (int k0 = 0; k0 < DIM; k0 += 4) {
        v2f a = *(const v2f*)(arow + k0);                 // A: global_load_b64
        const float* wrow = Wm + (size_t)(k0 + koffs) * DIM + lm;
#pragma unroll
        for (int t = 0; t < 8; ++t) {
            const int n0 = (w * 8 + t) * 16;
            v2f bv;
            bv.x = wrow[n0];          // W[k0+koffs  ][n0+lm]
            bv.y = wrow[DIM + n0];    // W[k0+koffs+1][n0+lm]
            acc[t] = __builtin_amdgcn_wmma_f32_16x16x4_f32(
                false, a, false, bv, (short)0, acc[t], false, false);
        }
    }

    // write X tile to LDS with +4*row swizzle (conflict-free stage-2 reads)
#pragma unroll
    for (int t = 0; t < 8; ++t) {
        const int n0 = (w * 8 + t) * 16;
#pragma unroll
        for (int r = 0; r < 8; ++r) {
            const int m = r + hi * 8;
            const int n = n0 + lm;
            Xs[m * DIM + ((n + 4 * m) & (DIM - 1))] = acc[t][r];
        }
    }
    __syncthreads();

    // -------- Stage 2: S(16x512) = Xs @ H2_b^T, fused epilogue --------
    v8f acc2[4] = {};
    const float* b2base[4];
#pragma unroll
    for (int t = 0; t < 4; ++t) {
        const int j0 = (w * 4 + t) * 16;
        b2base[t] = in2 + ((size_t)(j0 + lm) * BSZ + bb) * DIM + koffs;
    }

    for (int k0 = 0; k0 < DIM; k0 += 4) {
        const int kk = k0 + koffs;
        v2f a = *(const v2f*)(&Xs[lm * DIM + ((kk + 4 * lm) & (DIM - 1))]); // ds_load_b64
#pragma unroll
        for (int t = 0; t < 4; ++t) {
            v2f bv = *(const v2f*)(b2base[t] + k0);
            acc2[t] = __builtin_amdgcn_wmma_f32_16x16x4_f32(
                false, a, false, bv, (short)0, acc2[t], false, false);
        }
    }

    const float b0 = bias[0];
    int mok1[8];
#pragma unroll
    for (int r = 0; r < 8; ++r)
        mok1[r] = (m1[(i0 + r + hi * 8) * BSZ + bb] == 0);

#pragma unroll
    for (int t = 0; t < 4; ++t) {
        const int j = (w * 4 + t) * 16 + lm;
        const int mok2 = (m2[j * BSZ + bb] == 0);
#pragma unroll
        for (int r = 0; r < 8; ++r) {
            const int i = i0 + r + hi * 8;
            float s = acc2[t][r] + b0;
            s = fmaxf(s, 0.0f);                               // relu
            // order-preserving key; s >= 0 so positive mapping suffices.
            unsigned key = (mok1[r] && mok2) ? (__float_as_uint(s) | 0x80000000u) : 0u;
            keys[((size_t)(bb * LEN + i) << 9) + j] = key;
        }
    }
}

// ---------------------------------------------------------------------------
// Kernel 2: per-batch top-256 via 4-pass radix select + bitonic sort.
// grid = 32 batches, block = 512 threads.
// ---------------------------------------------------------------------------
__global__ __launch_bounds__(512) void topk_kernel(
    const unsigned* __restrict__ keys, float* __restrict__ out)
{
    const int bb  = blockIdx.x;
    const int tid = threadIdx.x;
    const int NT  = blockDim.x;
    const unsigned* kb = keys + (size_t)bb * LEN * LEN;
    const int N = LEN * LEN;

    __shared__ unsigned hist[256];
    __shared__ unsigned sh_prefix, sh_krem, sh_cnt;
    __shared__ unsigned cand[TOPK];
    __shared__ int sh_last;

    if (tid < 256) hist[tid] = 0u;
    if (tid == 0) { sh_prefix = 0u; sh_krem = TOPK; sh_cnt = 0u; sh_last = -1; }
    __syncthreads();

    // ---- radix select: find 256th-largest key T ----
    for (int pass = 0; pass < 4; ++pass) {
        const int shift = 24 - 8 * pass;
        const unsigned hi_mask = (pass == 0) ? 0u : (0xFFFFFFFFu << (shift + 8));
        const unsigned pfx = sh_prefix;
        for (int idx = tid; idx < N; idx += NT) {
            unsigned e = kb[idx];
            if ((e & hi_mask) == pfx)
                atomicAdd(&hist[(e >> shift) & 0xFFu], 1u);
        }
        __syncthreads();
        if (tid == 0) {
            unsigned kr = sh_krem;
            unsigned v = 255u;
            for (;; --v) {
                unsigned c = hist[v];
                if (kr <= c || v == 0u) break;
                kr -= c;
            }
            sh_prefix |= (v << shift);
            sh_krem = kr;
        }
        __syncthreads();
        if (tid < 256) hist[tid] = 0u;
        __syncthreads();
    }

    const unsigned T   = sh_prefix;
    const unsigned ngt = TOPK - sh_krem;   // #keys strictly greater than T

    // ---- gather keys > T, pad with T ----
    for (int idx = tid; idx < N; idx += NT) {
        unsigned e = kb[idx];
        if (e > T) {
            unsigned p = atomicAdd(&sh_cnt, 1u);
            if (p < TOPK) cand[p] = e;
        }
    }
    __syncthreads();
    if (tid >= (int)ngt && tid < TOPK) cand[tid] = T;
    __syncthreads();

    // ---- bitonic sort, descending ----
    for (int ksz = 2; ksz <= TOPK; ksz <<= 1) {
        for (int jsz = ksz >> 1; jsz > 0; jsz >>= 1) {
            if (tid < TOPK) {
                const int ixj = tid ^ jsz;
                if (ixj > tid) {
                    const bool desc = ((tid & ksz) == 0);
                    unsigned a = cand[tid], c = cand[ixj];
                    if (desc ? (a < c) : (a > c)) { cand[tid] = c; cand[ixj] = a; }
                }
            }
            __syncthreads();
        }
    }

    // ---- -inf back-fill rule + key -> float, write output ----
    if (tid < TOPK && cand[tid] != 0u) atomicMax(&sh_last, tid);
    __syncthreads();
    if (tid < TOPK) {
        unsigned key = cand[tid];
        if (key == 0u && sh_last >= 0) key = cand[sh_last];
        float val;
        if (key == 0u)                 val = __uint_as_float(0xFF800000u); // -inf
        else if (key & 0x80000000u)    val = __uint_as_float(key & 0x7FFFFFFFu);
        else                           val = __uint_as_float(~key);
        out[bb * TOPK + tid] = val;
    }
}

// ---------------------------------------------------------------------------
extern "C" void kernel_launch(void* const* d_in, const int* in_sizes, int n_in,
                              void* d_out, int out_size, void* d_ws, size_t ws_size,
                              hipStream_t stream) {
    const float* in1  = (const float*)d_in[0];   // inputs1 [512,32,1024]
    const float* in2  = (const float*)d_in[1];   // inputs2 [512,32,1024]
    const int*   m1   = (const int*)d_in[2];     // mask1   [512,32]
    const int*   m2   = (const int*)d_in[3];     // mask2   [512,32]
    const float* Wm   = (const float*)d_in[4];   // W       [1024,1024]
    const float* bias = (const float*)d_in[5];   // b       [1]
    unsigned* keys = (unsigned*)d_ws;            // 32*512*512*4 = 32 MB
    float* out = (float*)d_out;                  // [32, 256]

    bilinear_keys_kernel<<<dim3(LEN / 16, BSZ), 256, 0, stream>>>(
        in1, in2, m1, m2, Wm, bias, keys);
    topk_kernel<<<dim3(BSZ), 512, 0, stream>>>(keys, out);
}